// GATModel_53128745451631
// MI455X (gfx1250) — compile-verified
//
#include <hip/hip_runtime.h>
#include <math.h>

typedef __attribute__((ext_vector_type(2))) float v2f;
typedef __attribute__((ext_vector_type(8))) float v8f;

#define FIN   128
#define HEADS 4
#define CDIM  32
#define HOUT  128      // HEADS*CDIM
#define NEG_SLOPE 0.2f

// ---------- ordered-uint mapping for float atomic max (monotonic) ----------
__device__ __forceinline__ unsigned ordKey(float f) {
    unsigned u = __float_as_uint(f);
    return (u & 0x80000000u) ? ~u : (u | 0x80000000u);
}
__device__ __forceinline__ float ordInv(unsigned k) {
    unsigned u = (k & 0x80000000u) ? (k & 0x7FFFFFFFu) : ~k;
    return __uint_as_float(u);
}

// ---------- GEMM: H = act(X) @ W, X:[nRows,128], W:[128,128] row-major ----------
// One wave (32 threads) computes a 16x64 output strip with 4 f32 WMMA accumulators.
// grid = ((nRows+15)/16, 2), block = 32.
// Row index is clamped (not branched) so the k-loop has no EXEC manipulation;
// out-of-range rows compute garbage that is simply never stored.
template<int RELU>
__global__ __launch_bounds__(32)
void k_gemm_wmma(const float* __restrict__ X, const float* __restrict__ W,
                 float* __restrict__ H, int nRows) {
    const int lane   = threadIdx.x;            // 0..31
    const int tm     = blockIdx.x;             // 16-row tile
    const int tc4    = blockIdx.y;             // 64-col group
    const int mlane  = lane & 15;
    const int khalf  = (lane >> 4) * 2;        // 0 or 2
    int row          = tm * 16 + mlane;
    if (row >= nRows) row = nRows - 1;         // clamp: keeps loads in-bounds, no divergence
    const int colB   = tc4 * 64 + mlane;

    const float* xr = X + (size_t)row * FIN + khalf;

    v8f acc0 = {}, acc1 = {}, acc2 = {}, acc3 = {};

    for (int k0 = 0; k0 < FIN; k0 += 4) {
        v2f a;
        float a0 = xr[k0];
        float a1 = xr[k0 + 1];
        if (RELU) { a0 = fmaxf(a0, 0.0f); a1 = fmaxf(a1, 0.0f); }
        a.x = a0; a.y = a1;

        const float* w0 = W + (size_t)(k0 + khalf) * HOUT;
        const float* w1 = w0 + HOUT;
        v2f b0, b1, b2, b3;
        b0.x = w0[colB +  0]; b0.y = w1[colB +  0];
        b1.x = w0[colB + 16]; b1.y = w1[colB + 16];
        b2.x = w0[colB + 32]; b2.y = w1[colB + 32];
        b3.x = w0[colB + 48]; b3.y = w1[colB + 48];

        acc0 = __builtin_amdgcn_wmma_f32_16x16x4_f32(false, a, false, b0, (short)0, acc0, false, false);
        acc1 = __builtin_amdgcn_wmma_f32_16x16x4_f32(false, a, false, b1, (short)0, acc1, false, false);
        acc2 = __builtin_amdgcn_wmma_f32_16x16x4_f32(false, a, false, b2, (short)0, acc2, false, false);
        acc3 = __builtin_amdgcn_wmma_f32_16x16x4_f32(false, a, false, b3, (short)0, acc3, false, false);
    }

    // C/D layout: VGPR j, lanes 0-15 -> M=j, lanes 16-31 -> M=j+8; N = lane&15
    const int mOff = (lane >> 4) * 8;
    for (int j = 0; j < 8; ++j) {
        int r = tm * 16 + mOff + j;
        if (r < nRows) {
            float* hr = H + (size_t)r * HOUT + tc4 * 64 + mlane;
            hr[0]  = acc0[j];
            hr[16] = acc1[j];
            hr[32] = acc2[j];
            hr[48] = acc3[j];
        }
    }
}

// ---------- per-(node,head) attention coefficients ----------
__global__ void k_attn_coeff(const float* __restrict__ H,
                             const float* __restrict__ a_src,
                             const float* __restrict__ a_dst,
                             float* __restrict__ als, float* __restrict__ ald,
                             int n) {
    int i = blockIdx.x * blockDim.x + threadIdx.x;   // node*HEADS + h
    if (i >= n * HEADS) return;
    int node = i >> 2, h = i & 3;
    const float* hp = H + (size_t)node * HOUT + h * CDIM;
    const float* as = a_src + h * CDIM;
    const float* ad = a_dst + h * CDIM;
    float s = 0.0f, d = 0.0f;
#pragma unroll
    for (int c = 0; c < CDIM; ++c) { float v = hp[c]; s = fmaf(v, as[c], s); d = fmaf(v, ad[c], d); }
    als[i] = s; ald[i] = d;
}

// ---------- init: m = ordKey(-inf), denom = 0 ----------
__global__ void k_init_attn(unsigned* __restrict__ mkey, float* __restrict__ denom, int n) {
    int i = blockIdx.x * blockDim.x + threadIdx.x;
    if (i < n * HEADS) { mkey[i] = ordKey(-INFINITY); denom[i] = 0.0f; }
}

// ---------- init aggregation buffer with broadcast bias ----------
__global__ void k_init_out(float* __restrict__ out, const float* __restrict__ bias, int n) {
    int i = blockIdx.x * blockDim.x + threadIdx.x;
    if (i < n * HOUT) out[i] = bias[i & (HOUT - 1)];
}

// ---------- edge pass 1: segment max of leaky-relu attention logits ----------
__global__ void k_edge_max(const int* __restrict__ src, const int* __restrict__ dst,
                           const float* __restrict__ als, const float* __restrict__ ald,
                           unsigned* __restrict__ mkey, int e_total) {
    int i = blockIdx.x * blockDim.x + threadIdx.x;   // edge*HEADS + h
    if (i >= e_total * HEADS) return;
    int e = i >> 2, h = i & 3;
    int s = src[e], d = dst[e];
    float v = als[s * HEADS + h] + ald[d * HEADS + h];
    v = (v > 0.0f) ? v : v * NEG_SLOPE;
    atomicMax(&mkey[d * HEADS + h], ordKey(v));
}

// ---------- edge pass 2: w = exp(e - m[dst]); denom[dst] += w ----------
__global__ void k_edge_exp(const int* __restrict__ src, const int* __restrict__ dst,
                           const float* __restrict__ als, const float* __restrict__ ald,
                           const unsigned* __restrict__ mkey,
                           float* __restrict__ w, float* __restrict__ denom, int e_total) {
    int i = blockIdx.x * blockDim.x + threadIdx.x;
    if (i >= e_total * HEADS) return;
    int e = i >> 2, h = i & 3;
    int s = src[e], d = dst[e];
    float v = als[s * HEADS + h] + ald[d * HEADS + h];
    v = (v > 0.0f) ? v : v * NEG_SLOPE;
    float mm = ordInv(mkey[d * HEADS + h]);          // finite: dst has >=1 incoming edge
    float ww = expf(v - mm);
    w[i] = ww;
    __hip_atomic_fetch_add(&denom[d * HEADS + h], ww, __ATOMIC_RELAXED, __HIP_MEMORY_SCOPE_AGENT);
}

// ---------- edge pass 3: out[dst] += h[src] * alpha ----------
__global__ void k_edge_scatter(const int* __restrict__ src, const int* __restrict__ dst,
                               const float* __restrict__ Hm,
                               const float* __restrict__ w, const float* __restrict__ denom,
                               float* __restrict__ out, int e_total) {
    int i = blockIdx.x * blockDim.x + threadIdx.x;   // edge*HEADS + h
    if (i >= e_total * HEADS) return;
    int e = i >> 2, h = i & 3;
    int s = src[e], d = dst[e];
    float alpha = w[i] / fmaxf(denom[d * HEADS + h], 1e-16f);
    const float4* hs = (const float4*)(Hm + (size_t)s * HOUT + h * CDIM);
    float* op = out + (size_t)d * HOUT + h * CDIM;
#pragma unroll
    for (int c4 = 0; c4 < CDIM / 4; ++c4) {
        float4 hv = hs[c4];
        __hip_atomic_fetch_add(op + c4 * 4 + 0, hv.x * alpha, __ATOMIC_RELAXED, __HIP_MEMORY_SCOPE_AGENT);
        __hip_atomic_fetch_add(op + c4 * 4 + 1, hv.y * alpha, __ATOMIC_RELAXED, __HIP_MEMORY_SCOPE_AGENT);
        __hip_atomic_fetch_add(op + c4 * 4 + 2, hv.z * alpha, __ATOMIC_RELAXED, __HIP_MEMORY_SCOPE_AGENT);
        __hip_atomic_fetch_add(op + c4 * 4 + 3, hv.w * alpha, __ATOMIC_RELAXED, __HIP_MEMORY_SCOPE_AGENT);
    }
}

extern "C" void kernel_launch(void* const* d_in, const int* in_sizes, int n_in,
                              void* d_out, int out_size, void* d_ws, size_t ws_size,
                              hipStream_t stream) {
    const float* x      = (const float*)d_in[0];
    const int*   ei     = (const int*)d_in[1];
    const float* W1     = (const float*)d_in[2];
    const float* a_src1 = (const float*)d_in[3];
    const float* a_dst1 = (const float*)d_in[4];
    const float* b1     = (const float*)d_in[5];
    const float* W2     = (const float*)d_in[6];
    const float* a_src2 = (const float*)d_in[7];
    const float* a_dst2 = (const float*)d_in[8];
    const float* b2     = (const float*)d_in[9];

    const int N = in_sizes[0] / FIN;
    const int E = in_sizes[1] / 2;
    const int* src = ei;          // edge_index[0]
    const int* dst = ei + E;      // edge_index[1]

    // workspace layout (all 16B-aligned chunks), total ~83 MB
    char* p = (char*)d_ws;
    float*    H     = (float*)p;    p += (size_t)N * HOUT * sizeof(float);
    float*    als   = (float*)p;    p += (size_t)N * HEADS * sizeof(float);
    float*    ald   = (float*)p;    p += (size_t)N * HEADS * sizeof(float);
    unsigned* mkey  = (unsigned*)p; p += (size_t)N * HEADS * sizeof(unsigned);
    float*    denom = (float*)p;    p += (size_t)N * HEADS * sizeof(float);
    float*    w     = (float*)p;    p += (size_t)E * HEADS * sizeof(float);

    float* out = (float*)d_out;

    dim3 gGemm((N + 15) / 16, 2);
    const int BT = 256;
    int gNH  = (N * HEADS + BT - 1) / BT;
    int gNO  = (N * HOUT  + BT - 1) / BT;
    int gEH  = (E * HEADS + BT - 1) / BT;

    // ---------------- Layer 1 (aggregate into d_out) ----------------
    k_gemm_wmma<0><<<gGemm, 32, 0, stream>>>(x, W1, H, N);
    k_attn_coeff<<<gNH, BT, 0, stream>>>(H, a_src1, a_dst1, als, ald, N);
    k_init_attn <<<gNH, BT, 0, stream>>>(mkey, denom, N);
    k_init_out  <<<gNO, BT, 0, stream>>>(out, b1, N);
    k_edge_max  <<<gEH, BT, 0, stream>>>(src, dst, als, ald, mkey, E);
    k_edge_exp  <<<gEH, BT, 0, stream>>>(src, dst, als, ald, mkey, w, denom, E);
    k_edge_scatter<<<gEH, BT, 0, stream>>>(src, dst, H, w, denom, out, E);

    // ---------------- Layer 2 (reads relu(d_out), aggregates back into d_out) ----------------
    k_gemm_wmma<1><<<gGemm, 32, 0, stream>>>(out, W2, H, N);   // H2 = relu(h1) @ W2
    k_attn_coeff<<<gNH, BT, 0, stream>>>(H, a_src2, a_dst2, als, ald, N);
    k_init_attn <<<gNH, BT, 0, stream>>>(mkey, denom, N);
    k_init_out  <<<gNO, BT, 0, stream>>>(out, b2, N);          // safe: gemm above already consumed old d_out
    k_edge_max  <<<gEH, BT, 0, stream>>>(src, dst, als, ald, mkey, E);
    k_edge_exp  <<<gEH, BT, 0, stream>>>(src, dst, als, ald, mkey, w, denom, E);
    k_edge_scatter<<<gEH, BT, 0, stream>>>(src, dst, H, w, denom, out, E);
}